// SauvolaMultiWindow_25761213841920
// MI455X (gfx1250) — compile-verified
//
#include <hip/hip_runtime.h>

// Sauvola multi-window threshold via two WMMA-accelerated prefix-scan passes
// (building SATs of x and x^2) + one SAT-lookup pass producing all 5 windows.
//
// HBM traffic ~ 33.5MB read + 67MB SAT (stays in 192MB L2) + 168MB output
// => essentially output-store bound (~11us floor at 23.3 TB/s).
//
// WMMA use: 16-wide inclusive prefix sums = multiply by triangular ones
// matrix, done with V_WMMA_F32_16X16X4_F32 chained over K (4 chunks of K=4),
// carry propagated through the C operand. fp32 in/out, so no precision loss
// versus a scalar scan.

typedef __attribute__((ext_vector_type(2))) float v2f;
typedef __attribute__((ext_vector_type(8))) float v8f;

namespace {
constexpr int IMG_B = 8;
constexpr int IMG_H = 1024;
constexpr int IMG_W = 1024;
constexpr int NWIN  = 5;
constexpr size_t PLANE = (size_t)IMG_H * IMG_W;
}

// ---------------------------------------------------------------------------
// Pass A: per-row inclusive prefix sums of x and x^2.
// One wave handles 16 rows x 1024 cols, sweeping 64 column tiles and carrying
// per-row running sums. Within a tile: D = X(16x16) * U(16x16, upper-tri ones)
// as 4 chained 16x16x4 WMMAs; carry enters via C, exits from column 15.
//
// fp32 16x16x4 VGPR layouts (ISA 7.12.2):
//   A 16x4 : lanes 0-15 & 16-31 both hold M=0..15; VGPR r holds K = r + 2*(lane>=16)
//   B 4x16 : lanes hold N=lane%16;                VGPR r holds K = r + 2*(lane>=16)
//   C/D    : lanes hold N=lane%16;                VGPR i holds M = i + 8*(lane>=16)
// ---------------------------------------------------------------------------
__global__ __launch_bounds__(32) void hprefix_wmma(
    const float* __restrict__ x, float* __restrict__ P1, float* __restrict__ P2)
{
  const int lane = threadIdx.x;
  const int half = lane >> 4;       // 0: lanes 0-15, 1: lanes 16-31
  const int m    = lane & 15;       // A row index; also B/D column index N
  const int kk0  = half << 1;       // K base for this half per VGPR layout

  const int img  = blockIdx.x >> 6;          // 8 images * 64 row-groups
  const int row0 = (blockIdx.x & 63) << 4;

  const float* xr = x + (size_t)img * PLANE + (size_t)(row0 + m) * IMG_W;
  float* p1 = P1 + (size_t)img * PLANE;
  float* p2 = P2 + (size_t)img * PLANE;

  float carry1[8], carry2[8];
#pragma unroll
  for (int i = 0; i < 8; ++i) { carry1[i] = 0.f; carry2[i] = 0.f; }

  for (int tile = 0; tile < IMG_W / 16; ++tile) {
    const int col0 = tile << 4;
    v8f c1, c2;
#pragma unroll
    for (int i = 0; i < 8; ++i) { c1[i] = carry1[i]; c2[i] = carry2[i]; }

#pragma unroll
    for (int ch = 0; ch < 4; ++ch) {
      // Lane needs X[m][4*ch + kk0 .. +1] -> one aligned float2 load.
      const float2 f = *(const float2*)(xr + col0 + (ch << 2) + kk0);
      v2f a1; a1[0] = f.x;       a1[1] = f.y;
      v2f a2; a2[0] = f.x * f.x; a2[1] = f.y * f.y;
      v2f b;
#pragma unroll
      for (int r = 0; r < 2; ++r) {
        const int kg = (ch << 2) + kk0 + r;
        b[r] = (kg <= m) ? 1.0f : 0.0f;      // U[k][n] = (k <= n); n == m per lane
      }
      c1 = __builtin_amdgcn_wmma_f32_16x16x4_f32(false, a1, false, b, (short)0, c1, false, false);
      c2 = __builtin_amdgcn_wmma_f32_16x16x4_f32(false, a2, false, b, (short)0, c2, false, false);
    }

#pragma unroll
    for (int i = 0; i < 8; ++i) {
      // Row total sits at N=15 (lane 15 / lane 31) -> width-16 broadcast.
      carry1[i] = __shfl(c1[i], 15, 16);
      carry2[i] = __shfl(c2[i], 15, 16);
      const size_t o = (size_t)(row0 + i + (half << 3)) * IMG_W + col0 + m;
      p1[o] = c1[i];
      p2[o] = c2[i];
    }
  }
}

// ---------------------------------------------------------------------------
// Pass B: in-place vertical prefix down each column group (turns row-prefix
// arrays into full SATs). D = L(16x16, lower-tri ones) * X(16x16); constant A,
// data in B; per-column carry via C, exits from row 15 (lane 16+n, VGPR 7).
// ---------------------------------------------------------------------------
__global__ __launch_bounds__(32) void vprefix_wmma(
    float* __restrict__ P1, float* __restrict__ P2)
{
  const int lane = threadIdx.x;
  const int half = lane >> 4;
  const int n    = lane & 15;       // column within tile; also A row index M
  const int kk0  = half << 1;

  const int plane = blockIdx.x & 1;            // 0 -> P1, 1 -> P2
  const int rest  = blockIdx.x >> 1;
  const int img   = rest >> 6;
  const int col0  = (rest & 63) << 4;

  float* P = (plane ? P2 : P1) + (size_t)img * PLANE;

  float carry = 0.0f;
  for (int tile = 0; tile < IMG_H / 16; ++tile) {
    const int row0 = tile << 4;
    v8f c;
#pragma unroll
    for (int i = 0; i < 8; ++i) c[i] = carry;   // same carry for all M of column n

#pragma unroll
    for (int ch = 0; ch < 4; ++ch) {
      v2f a, b;
#pragma unroll
      for (int r = 0; r < 2; ++r) {
        const int kg = (ch << 2) + kk0 + r;
        a[r] = (kg <= n) ? 1.0f : 0.0f;        // L[m][k] = (k <= m); m == n per lane
        b[r] = P[(size_t)(row0 + kg) * IMG_W + col0 + n];  // 64B-coalesced per half
      }
      c = __builtin_amdgcn_wmma_f32_16x16x4_f32(false, a, false, b, (short)0, c, false, false);
    }

    carry = __shfl(c[7], 16 + n, 32);           // D[15][n] from lane 16+n, VGPR 7
#pragma unroll
    for (int i = 0; i < 8; ++i)
      P[(size_t)(row0 + i + (half << 3)) * IMG_W + col0 + n] = c[i];
  }
}

// ---------------------------------------------------------------------------
// Pass C: SAT lookups (L2-resident) + Sauvola math; streams the 168MB output.
// Zero-padding border correction == analytic count of the clamped window box.
// ---------------------------------------------------------------------------
__global__ __launch_bounds__(256) void sauvola_out(
    const float* __restrict__ S1, const float* __restrict__ S2,
    const float* __restrict__ kv, const float* __restrict__ Rv,
    float* __restrict__ out)
{
  const int idx = blockIdx.x * 256 + threadIdx.x;   // < 8*1024*1024
  const int xc  = idx & (IMG_W - 1);
  const int yc  = (idx >> 10) & (IMG_H - 1);
  const int img = idx >> 20;

  const float* s1 = S1 + (size_t)img * PLANE;
  const float* s2 = S2 + (size_t)img * PLANE;

  const int radii[NWIN] = {3, 7, 15, 31, 63};
#pragma unroll
  for (int w = 0; w < NWIN; ++w) {
    const int r  = radii[w];
    const int y1 = max(yc - r, 0), y2 = min(yc + r, IMG_H - 1);
    const int x1 = max(xc - r, 0), x2 = min(xc + r, IMG_W - 1);
    // Branchless border handling: clamp index, zero the weight.
    const size_t rB = (size_t)y2 * IMG_W;
    const size_t rA = (size_t)max(y1 - 1, 0) * IMG_W;
    const int    xs = max(x1 - 1, 0);
    const float  wy = (y1 > 0) ? 1.0f : 0.0f;
    const float  wx = (x1 > 0) ? 1.0f : 0.0f;

    const float sum1 = s1[rB + x2] - wy * s1[rA + x2] - wx * s1[rB + xs] + wy * wx * s1[rA + xs];
    const float sum2 = s2[rB + x2] - wy * s2[rA + x2] - wx * s2[rB + xs] + wy * wx * s2[rA + xs];

    const float cnt = (float)((y2 - y1 + 1) * (x2 - x1 + 1));
    const float inv = __frcp_rn(cnt);
    const float E   = sum1 * inv;
    const float E2  = sum2 * inv;
    const float dev = __fsqrt_rn(fmaxf(E2 - E * E, 1e-6f));
    const float T   = E * (1.0f + kv[w] * (dev * __frcp_rn(Rv[w]) - 1.0f));

    out[((size_t)(img * NWIN + w)) * PLANE + (size_t)yc * IMG_W + xc] = T;
  }
}

// ---------------------------------------------------------------------------
extern "C" void kernel_launch(void* const* d_in, const int* in_sizes, int n_in,
                              void* d_out, int out_size, void* d_ws, size_t ws_size,
                              hipStream_t stream)
{
  const float* x  = (const float*)d_in[0];   // [8,1024,1024,1] fp32
  const float* kv = (const float*)d_in[1];   // [1,5,1,1,1]
  const float* Rv = (const float*)d_in[2];   // [1,5,1,1,1]
  float* out = (float*)d_out;                // [8,5,1024,1024,1] fp32

  // Workspace: two fp32 SAT planes, 32 MiB each (needs ws_size >= 64 MiB).
  float* P1 = (float*)d_ws;
  float* P2 = P1 + (size_t)IMG_B * PLANE;

  hprefix_wmma<<<IMG_B * (IMG_H / 16), 32, 0, stream>>>(x, P1, P2);
  vprefix_wmma<<<IMG_B * (IMG_W / 16) * 2, 32, 0, stream>>>(P1, P2);
  sauvola_out<<<(IMG_B * IMG_H * IMG_W) / 256, 256, 0, stream>>>(P1, P2, kv, Rv, out);

  (void)in_sizes; (void)n_in; (void)out_size; (void)ws_size;
}